// FocalLossWithDice_81406810128870
// MI455X (gfx1250) — compile-verified
//
#include <hip/hip_runtime.h>
#include <hip/hip_bf16.h>

// ---------------------------------------------------------------------------
// FocalLossWithDice for MI455X (gfx1250).
// Memory-bound: ~96 MiB/pass @ 23.3 TB/s => two streaming passes, B128
// vectorized loads (4 pixels/thread/iter), histogram top-k selection,
// WMMA (v_wmma_f32_16x16x4_f32) wave reductions, ballot-aggregated atomics.
// ---------------------------------------------------------------------------

#define NCH     5
#define HW      (1u << 20)          // 1024*1024
#define NPIX    (4u << 20)          // B*H*W = 4,194,304
#define HWV     (1u << 18)          // HW / 4 (float4 units)
#define NVEC    (1u << 20)          // NPIX / 4 (float4 groups)
#define NCLS    4                   // dice classes 1..4
#define NBINS   2048                // |dt-do| histogram bins over [0,1]
#define TOPK    16384
#define EPS_MC  1e-5f
#define EPS_BIN 1e-3f
#define MIN_PIX 10.0f

#define NBLOCKS 1024
#define NTHREADS 256                // 8 wave32/block; NVEC = 4 * (1024*256)

struct Ws {
    float    scal[4];               // ce_sum, bin_inter, bin_osum, bin_tcnt
    float    cls[NCLS][3];          // inter, dos(sum p), dts(sum dt) over top-k
    int      thrT[NCLS];            // threshold bin per class
    int      thrNeed[NCLS];         // tie-bin admission budget
    unsigned ticket[NCLS];          // tie-bin admission counter
    unsigned hist[NCLS][NBINS];     // global histogram
};

typedef float v2f __attribute__((ext_vector_type(2)));
typedef float v8f __attribute__((ext_vector_type(8)));

// Wave32 sum on the matrix pipe: A carries the 32 lane partials, B = ones.
// D[m][n] = rowsum(A)[m]; rows 0-7 live in lanes 0-15, rows 8-15 in lanes
// 16-31, so sum-of-8-D-VGPRs + one xor-16 shuffle yields the full wave sum.
// EXEC must be all ones at the call site.
__device__ __forceinline__ float wave_sum32(float v) {
#if __has_builtin(__builtin_amdgcn_wmma_f32_16x16x4_f32)
    v2f a; a[0] = v;    a[1] = 0.0f;
    v2f b; b[0] = 1.0f; b[1] = 1.0f;
    v8f c = {};
    v8f d = __builtin_amdgcn_wmma_f32_16x16x4_f32(
        false, a, false, b, (short)0, c, false, false);
    float s = ((d[0] + d[1]) + (d[2] + d[3])) + ((d[4] + d[5]) + (d[6] + d[7]));
    s += __shfl_xor(s, 16, 32);
    return s;
#else
    for (int off = 16; off > 0; off >>= 1) v += __shfl_xor(v, off, 32);
    return v;
#endif
}

__device__ __forceinline__ float f4_at(const float4& v, int j) {
    return (j == 0) ? v.x : (j == 1) ? v.y : (j == 2) ? v.z : v.w;
}
__device__ __forceinline__ int i4_at(const int4& v, int j) {
    return (j == 0) ? v.x : (j == 1) ? v.y : (j == 2) ? v.z : v.w;
}

// ---------------------------------------------------------------------------
// Pass 1: fused softmax -> CE sum, binary-dice sums, per-class |dt-p| hist.
// ---------------------------------------------------------------------------
__global__ void __launch_bounds__(NTHREADS)
focal_dice_pass1(const float* __restrict__ outputs,
                 const int*   __restrict__ targets,
                 Ws* __restrict__ ws) {
    __shared__ unsigned sh_hist[NCLS][NBINS];      // 32 KiB
    __shared__ float    sh_red[4];

    const int tid = threadIdx.x;
    for (int i = tid; i < NCLS * NBINS; i += blockDim.x)
        ((unsigned*)sh_hist)[i] = 0u;
    if (tid < 4) sh_red[tid] = 0.0f;
    __syncthreads();

    float ce = 0.f, bi = 0.f, bo = 0.f, bt = 0.f;

    const float4* __restrict__ out4 = (const float4*)outputs;
    const int4*   __restrict__ tgt4 = (const int4*)targets;

    const unsigned stride = gridDim.x * blockDim.x;        // 262,144
    for (unsigned g = blockIdx.x * blockDim.x + tid; g < NVEC; g += stride) {
        const unsigned b  = g >> 18;
        const unsigned iv = g & (HWV - 1u);
        const float4* chan = out4 + (((size_t)b * NCH) << 18) + iv;

        // prime L2 for the next grid-stride iteration (speculative prefetch)
        __builtin_prefetch(chan + stride, 0, 0);
        __builtin_prefetch(chan + ((size_t)4 << 18) + stride, 0, 0);
        __builtin_prefetch(tgt4 + g + stride, 0, 0);

        const float4 c0 = chan[0];                         // global_load_b128 x5
        const float4 c1 = chan[(size_t)1 << 18];
        const float4 c2 = chan[(size_t)2 << 18];
        const float4 c3 = chan[(size_t)3 << 18];
        const float4 c4 = chan[(size_t)4 << 18];
        const int4   tv = tgt4[g];

        #pragma unroll
        for (int j = 0; j < 4; ++j) {
            const float x0 = f4_at(c0, j), x1 = f4_at(c1, j), x2 = f4_at(c2, j);
            const float x3 = f4_at(c3, j), x4 = f4_at(c4, j);
            const int   t  = i4_at(tv, j);

            const float m  = fmaxf(fmaxf(fmaxf(x0, x1), fmaxf(x2, x3)), x4);
            const float e0 = __expf(x0 - m), e1 = __expf(x1 - m), e2 = __expf(x2 - m);
            const float e3 = __expf(x3 - m), e4 = __expf(x4 - m);
            const float S    = ((e0 + e1) + (e2 + e3)) + e4;
            const float inv  = 1.0f / S;
            const float logS = __logf(S);

            const float p0 = e0 * inv, p1 = e1 * inv, p2 = e2 * inv;
            const float p3 = e3 * inv, p4 = e4 * inv;

            // focal CE at target class
            const float xt = (t == 0) ? x0 : (t == 1) ? x1 : (t == 2) ? x2 : (t == 3) ? x3 : x4;
            const float pt = (t == 0) ? p0 : (t == 1) ? p1 : (t == 2) ? p2 : (t == 3) ? p3 : p4;
            const float logpt = (xt - m) - logS;
            const float om = 1.0f - pt;
            ce += om * om * logpt;                 // negated at finalize

            // binary localization dice terms
            const float o = 1.0f - p0;
            bo += o;
            if (t > 0) { bi += o; bt += 1.0f; }

            // per-class hardness histograms (classes 1..4)
            #pragma unroll
            for (int c = 1; c <= NCLS; ++c) {
                const float pc = (c == 1) ? p1 : (c == 2) ? p2 : (c == 3) ? p3 : p4;
                const float dt = (t == c) ? 1.0f : 0.0f;
                const float lb = fabsf(dt - pc);
                int bin = (int)(lb * (float)NBINS);
                bin = (bin > NBINS - 1) ? (NBINS - 1) : bin;
                atomicAdd(&sh_hist[c - 1][bin], 1u);
            }
        }
    }

    // wave reduction on the matrix pipe (EXEC all ones: uniform trip count)
    ce = wave_sum32(ce);
    bi = wave_sum32(bi);
    bo = wave_sum32(bo);
    bt = wave_sum32(bt);
    if ((tid & 31) == 0) {
        atomicAdd(&sh_red[0], ce);
        atomicAdd(&sh_red[1], bi);
        atomicAdd(&sh_red[2], bo);
        atomicAdd(&sh_red[3], bt);
    }
    __syncthreads();
    if (tid < 4) atomicAdd(&ws->scal[tid], sh_red[tid]);

    // flush only non-zero histogram bins
    for (int i = tid; i < NCLS * NBINS; i += blockDim.x) {
        const unsigned v = ((unsigned*)sh_hist)[i];
        if (v) atomicAdd(&((unsigned*)ws->hist)[i], v);
    }
}

// ---------------------------------------------------------------------------
// Threshold finder: per class, largest bin T with count(bins >= T) >= K.
// ---------------------------------------------------------------------------
__global__ void focal_dice_find_threshold(Ws* __restrict__ ws) {
    const int c = threadIdx.x;
    if (c >= NCLS || blockIdx.x != 0) return;
    unsigned cum = 0, above = 0;
    int T = -1;
    for (int b = NBINS - 1; b >= 0; --b) {
        const unsigned h = ws->hist[c][b];
        if (T < 0 && (cum + h) >= (unsigned)TOPK) { T = b; above = cum; }
        cum += h;
    }
    if (T < 0) { T = 0; above = cum - ws->hist[c][0]; }  // N < K: cannot happen
    ws->thrT[c]    = T;
    ws->thrNeed[c] = (int)((unsigned)TOPK - above);
}

// ---------------------------------------------------------------------------
// Pass 2: recompute softmax, accumulate top-k dice sums per class.
// Strictly-above-threshold elements always count; tie-bin elements are
// admitted via a ballot-aggregated atomic ticket (one atomic per wave).
// ---------------------------------------------------------------------------
__global__ void __launch_bounds__(NTHREADS)
focal_dice_pass2(const float* __restrict__ outputs,
                 const int*   __restrict__ targets,
                 Ws* __restrict__ ws) {
    const int tid  = threadIdx.x;
    const int lane = tid & 31;

    int T[NCLS], need[NCLS];
    #pragma unroll
    for (int c = 0; c < NCLS; ++c) { T[c] = ws->thrT[c]; need[c] = ws->thrNeed[c]; }

    float inter[NCLS] = {}, dos_[NCLS] = {}, dts_[NCLS] = {};

    const float4* __restrict__ out4 = (const float4*)outputs;
    const int4*   __restrict__ tgt4 = (const int4*)targets;

    const unsigned stride = gridDim.x * blockDim.x;
    for (unsigned g = blockIdx.x * blockDim.x + tid; g < NVEC; g += stride) {
        const unsigned b  = g >> 18;
        const unsigned iv = g & (HWV - 1u);
        const float4* chan = out4 + (((size_t)b * NCH) << 18) + iv;

        __builtin_prefetch(chan + stride, 0, 0);
        __builtin_prefetch(chan + ((size_t)4 << 18) + stride, 0, 0);
        __builtin_prefetch(tgt4 + g + stride, 0, 0);

        const float4 c0 = chan[0];
        const float4 c1 = chan[(size_t)1 << 18];
        const float4 c2 = chan[(size_t)2 << 18];
        const float4 c3 = chan[(size_t)3 << 18];
        const float4 c4 = chan[(size_t)4 << 18];
        const int4   tv = tgt4[g];

        #pragma unroll
        for (int j = 0; j < 4; ++j) {
            const float x0 = f4_at(c0, j), x1 = f4_at(c1, j), x2 = f4_at(c2, j);
            const float x3 = f4_at(c3, j), x4 = f4_at(c4, j);
            const int   t  = i4_at(tv, j);

            const float m  = fmaxf(fmaxf(fmaxf(x0, x1), fmaxf(x2, x3)), x4);
            const float e0 = __expf(x0 - m), e1 = __expf(x1 - m), e2 = __expf(x2 - m);
            const float e3 = __expf(x3 - m), e4 = __expf(x4 - m);
            const float inv = 1.0f / (((e0 + e1) + (e2 + e3)) + e4);
            const float pr[NCLS] = { e1 * inv, e2 * inv, e3 * inv, e4 * inv };

            #pragma unroll
            for (int c = 0; c < NCLS; ++c) {
                const float dt = (t == c + 1) ? 1.0f : 0.0f;
                const float pc = pr[c];
                const float lb = fabsf(dt - pc);
                int bin = (int)(lb * (float)NBINS);
                bin = (bin > NBINS - 1) ? (NBINS - 1) : bin;

                bool take = (bin > T[c]);
                const bool want = (bin == T[c]);

                // wave-aggregated tie ticket: one global atomic per wave
                const unsigned mask = (unsigned)__ballot(want);
                if (mask) {
                    const int leader = __ffs(mask) - 1;
                    unsigned base = 0;
                    if (lane == leader)
                        base = atomicAdd(&ws->ticket[c], (unsigned)__popc(mask));
                    base = (unsigned)__shfl((int)base, leader, 32);
                    const unsigned rank = (unsigned)__popc(mask & ((1u << lane) - 1u));
                    if (want && (base + rank) < (unsigned)need[c]) take = true;
                }
                if (take) { inter[c] += pc * dt; dos_[c] += pc; dts_[c] += dt; }
            }
        }
    }

    // 12 WMMA wave reductions + global atomics (EXEC all ones here)
    #pragma unroll
    for (int c = 0; c < NCLS; ++c) {
        const float s0 = wave_sum32(inter[c]);
        const float s1 = wave_sum32(dos_[c]);
        const float s2 = wave_sum32(dts_[c]);
        if (lane == 0) {
            atomicAdd(&ws->cls[c][0], s0);
            atomicAdd(&ws->cls[c][1], s1);
            atomicAdd(&ws->cls[c][2], s2);
        }
    }
}

// ---------------------------------------------------------------------------
// Finalize: combine the three loss terms into the scalar output.
// ---------------------------------------------------------------------------
__global__ void focal_dice_finalize(const Ws* __restrict__ ws, float* __restrict__ out) {
    if (threadIdx.x != 0 || blockIdx.x != 0) return;

    const float ce = -ws->scal[0] / (float)NPIX;

    float d = 0.0f;
    #pragma unroll
    for (int c = 0; c < NCLS; ++c) {
        const float inter = ws->cls[c][0];
        const float dos   = ws->cls[c][1];
        const float dts   = ws->cls[c][2];
        const float uni   = dos + dts + EPS_MC;
        const float l     = 1.0f - (2.0f * inter + EPS_MC) / uni;
        d += (dts > MIN_PIX) ? l : 0.0f;
    }
    d *= (1.0f / (float)NCLS);

    const float uni = ws->scal[2] + ws->scal[3] + EPS_BIN;
    const float loc = 1.0f - (2.0f * ws->scal[1] + EPS_BIN) / uni;

    out[0] = ce + 0.1f * d + 0.1f * loc;
}

// ---------------------------------------------------------------------------
extern "C" void kernel_launch(void* const* d_in, const int* in_sizes, int n_in,
                              void* d_out, int out_size, void* d_ws, size_t ws_size,
                              hipStream_t stream) {
    const float* outputs = (const float*)d_in[0];
    const int*   targets = (const int*)d_in[1];
    float*       out     = (float*)d_out;
    Ws*          ws      = (Ws*)d_ws;

    (void)in_sizes; (void)n_in; (void)out_size; (void)ws_size;

    // Deterministic per-call scratch init (graph-capture-safe memset node).
    hipMemsetAsync(d_ws, 0, sizeof(Ws), stream);

    focal_dice_pass1<<<NBLOCKS, NTHREADS, 0, stream>>>(outputs, targets, ws);
    focal_dice_find_threshold<<<1, 64, 0, stream>>>(ws);
    focal_dice_pass2<<<NBLOCKS, NTHREADS, 0, stream>>>(outputs, targets, ws);
    focal_dice_finalize<<<1, 64, 0, stream>>>(ws, out);
}